// FusedTensorProductScatter_60919816127117
// MI455X (gfx1250) — compile-verified
//
#include <hip/hip_runtime.h>
#include <cstdint>

typedef __attribute__((ext_vector_type(2))) float v2f;
typedef __attribute__((ext_vector_type(8))) float v8f;

#define MULT 32
#define IN_DIM 128           // 32 scalars + 32*3 vector components
#define W_PER_EDGE 4096      // 4 * 32 * 32 floats per edge
#define WAVES_PER_BLOCK 8

__constant__ const float kInvSqrt3 = 0.57735026918962576451f;
__constant__ const float kPathNorm = 0.125f;   // 1/sqrt(2*32)

__global__ void zero_out_kernel(float* __restrict__ out, int n) {
    int i = blockIdx.x * blockDim.x + threadIdx.x;
    if (i < n) out[i] = 0.0f;
}

__global__ __launch_bounds__(256)
void FusedTensorProductScatter_kernel(const float* __restrict__ x,
                                      const float* __restrict__ edge_attr,
                                      const float* __restrict__ ew,
                                      const long long* __restrict__ esrc,
                                      const long long* __restrict__ edst,
                                      float* __restrict__ out,
                                      int E) {
    // Per-wave LDS: B operand (32 K x 16 N) + raw result columns 0..4 (5 x 128)
    __shared__ float sB[WAVES_PER_BLOCK][MULT][16];
    __shared__ float sR[WAVES_PER_BLOCK][5][IN_DIM];

    const int wave = threadIdx.x >> 5;
    const int lane = threadIdx.x & 31;
    const int e    = blockIdx.x * WAVES_PER_BLOCK + wave;
    if (e >= E) return;                       // wave-uniform: EXEC stays all-ones

    const float* __restrict__ W = ew + (size_t)e * W_PER_EDGE;
    __builtin_prefetch(W, 0, 0);              // global_prefetch_b8: kick the stream

    const long long src = esrc[e];
    const long long dst = edst[e];
    const float* __restrict__ xrow = x + (size_t)src * IN_DIM;

    const float e0  = edge_attr[4 * e + 0];
    const float e1a = edge_attr[4 * e + 1];
    const float e1b = edge_attr[4 * e + 2];
    const float e1c = edge_attr[4 * e + 3];

    // Lane u = lane holds xs0[u] and xs1[u, 0..2]
    const float xs0 = xrow[lane];
    const float x1a = xrow[MULT + 3 * lane + 0];
    const float x1b = xrow[MULT + 3 * lane + 1];
    const float x1c = xrow[MULT + 3 * lane + 2];
    const float y   = x1a * e1a + x1b * e1b + x1c * e1c;   // xs1 . e1

    // ---- Stage B (K=32, N=16): cols = [xs0, y, xs1_0, xs1_1, xs1_2, 0...] ----
    float (*B)[16] = sB[wave];
    B[lane][0] = xs0;
    B[lane][1] = y;
    B[lane][2] = x1a;
    B[lane][3] = x1b;
    B[lane][4] = x1c;
#pragma unroll
    for (int n = 5; n < 16; ++n) B[lane][n] = 0.0f;

    const int half = lane >> 4;    // 0: lanes 0-15 (K0/K1), 1: lanes 16-31 (K2/K3)
    const int l15  = lane & 15;

    // B fragments for all 8 K-steps (reused by every M-tile)
    v2f bfrag[8];
#pragma unroll
    for (int kt = 0; kt < 8; ++kt) {
        const int u0 = 4 * kt + 2 * half;
        bfrag[kt].x = B[u0 + 0][l15];
        bfrag[kt].y = B[u0 + 1][l15];
    }

    // ---- D = A' x B via 8 M-tiles x 8 K-steps of V_WMMA_F32_16X16X4_F32 ----
    // A'[32m + w, u] = w_m[u, w]  (m: 0=w000 1=w110 2=w011 3=w101)
    float (*R)[IN_DIM] = sR[wave];
#pragma unroll
    for (int mt = 0; mt < 8; ++mt) {
        const int m = mt >> 1;
        const int w = ((mt & 1) << 4) + l15;
        const float* __restrict__ Wm = W + m * (MULT * MULT) + w;
        v8f acc = {};
#pragma unroll
        for (int kt = 0; kt < 8; ++kt) {
            const int u0 = 4 * kt + 2 * half;
            v2f a;
            a.x = Wm[(u0 + 0) * MULT];   // A layout: lanes 0-15 K0/K1, 16-31 K2/K3
            a.y = Wm[(u0 + 1) * MULT];
            acc = __builtin_amdgcn_wmma_f32_16x16x4_f32(
                false, a, false, bfrag[kt], (short)0, acc, false, false);
        }
        // D layout: lane n = col, VGPR r = row (lanes 16-31: row+8). Keep cols 0..4.
        if (l15 < 5) {
            const int rbase = 16 * mt + 8 * half;
#pragma unroll
            for (int r = 0; r < 8; ++r) R[l15][rbase + r] = acc[r];
        }
    }

    // ---- Combine with e0/e1, nan-guard, scatter-add (4 outputs per lane) ----
    // v000[w]=R[0][w]  v110[w]=R[1][32+w]  v011[w]=R[0][64+w]  V101[w][k]=R[2+k][96+w]
    float* __restrict__ orow = out + (size_t)dst * IN_DIM;
#pragma unroll
    for (int t = 0; t < 4; ++t) {
        const int j = t * 32 + lane;
        float v;
        if (j < MULT) {
            v = kPathNorm * (R[0][j] * e0 + kInvSqrt3 * R[1][MULT + j]);
        } else {
            const int q = j - MULT;
            const int w = q / 3;
            const int k = q - 3 * w;
            const float ek = (k == 0) ? e1a : ((k == 1) ? e1b : e1c);
            v = kPathNorm * kInvSqrt3 * (R[0][64 + w] * ek + R[2 + k][96 + w] * e0);
        }
        if (!__builtin_isfinite(v)) v = 0.0f;
        atomicAdd(orow + j, v);
    }
}

extern "C" void kernel_launch(void* const* d_in, const int* in_sizes, int n_in,
                              void* d_out, int out_size, void* d_ws, size_t ws_size,
                              hipStream_t stream) {
    const float*     x         = (const float*)d_in[0];
    const float*     edge_attr = (const float*)d_in[1];
    const float*     ew        = (const float*)d_in[2];
    const long long* esrc      = (const long long*)d_in[3];
    const long long* edst      = (const long long*)d_in[4];
    float*           out       = (float*)d_out;

    const int E = in_sizes[3];   // edge_src element count

    const int zblocks = (out_size + 255) / 256;
    zero_out_kernel<<<zblocks, 256, 0, stream>>>(out, out_size);

    const int blocks = (E + WAVES_PER_BLOCK - 1) / WAVES_PER_BLOCK;
    FusedTensorProductScatter_kernel<<<blocks, 256, 0, stream>>>(
        x, edge_attr, ew, esrc, edst, out, E);
}